// PBELoss_171798691912
// MI455X (gfx1250) — compile-verified
//
#include <hip/hip_runtime.h>
#include <math.h>

// Node feature layout: [Pd, Qd, Pg, Qg, Vm, Va]
typedef float      f32x2 __attribute__((ext_vector_type(2)));
typedef float      f32x4 __attribute__((ext_vector_type(4)));
typedef long long  i64x2 __attribute__((ext_vector_type(2)));

#define THREADS 256

// ---------------------------------------------------------------------------
// Kernel 2a (defined FIRST so the disasm snippet shows it): edge message +
// segment-sum, 2 edges/thread (b128 edge streams).
//   I[dst] += (G + jB) * V[src]
// Streams are NT + prefetched; V gather is one b64/edge; atomic pair shares
// a cacheline. Hardware f32 atomics (global_atomic_add_f32) via
// unsafeAtomicAdd.
// ---------------------------------------------------------------------------
__global__ void pbe_edge_pair(const f32x4* __restrict__ attr2,   // (G0,B0,G1,B1)
                              const i64x2* __restrict__ src2,
                              const i64x2* __restrict__ dst2,
                              const f32x2* __restrict__ Vc,
                              float* __restrict__ Ic,
                              int nPairs) {
    int p = blockIdx.x * blockDim.x + threadIdx.x;
    if (p >= nPairs) return;

    // speculative prefetch 4096 pairs (128 waves) ahead on all three edge streams
    __builtin_prefetch(&attr2[p + 4096], 0, 1);
    __builtin_prefetch(&src2[p + 4096], 0, 1);
    __builtin_prefetch(&dst2[p + 4096], 0, 1);

    i64x2 s = __builtin_nontemporal_load(&src2[p]);
    i64x2 d = __builtin_nontemporal_load(&dst2[p]);
    f32x4 gb = __builtin_nontemporal_load(&attr2[p]);

    f32x2 v0 = Vc[s.x];   // single b64 gather per edge (L2-resident)
    f32x2 v1 = Vc[s.y];

    float mr0 = gb.x * v0.x - gb.y * v0.y;
    float mi0 = gb.x * v0.y + gb.y * v0.x;
    float mr1 = gb.z * v1.x - gb.w * v1.y;
    float mi1 = gb.z * v1.y + gb.w * v1.x;

    unsafeAtomicAdd(&Ic[2 * d.x + 0], mr0);
    unsafeAtomicAdd(&Ic[2 * d.x + 1], mi0);
    unsafeAtomicAdd(&Ic[2 * d.y + 0], mr1);
    unsafeAtomicAdd(&Ic[2 * d.y + 1], mi1);
}

// ---------------------------------------------------------------------------
// Kernel 1: per-node prep.
//   x = mask ? pred : target
//   Vc[i] = (Vm cos Va, Vm sin Va)      (interleaved complex)
//   Sc[i] = (Pg - Pd, Qg - Qd)          (interleaved complex)
// ---------------------------------------------------------------------------
__global__ void pbe_node_prep(const float* __restrict__ pred,
                              const float* __restrict__ target,
                              const unsigned char* __restrict__ mask,
                              f32x2* __restrict__ Vc,
                              f32x2* __restrict__ Sc,
                              int N) {
    int i = blockIdx.x * blockDim.x + threadIdx.x;
    if (i >= N) return;
    const int b = i * 6;
    float x[6];
#pragma unroll
    for (int k = 0; k < 6; ++k) {
        float p = __builtin_nontemporal_load(&pred[b + k]);
        float t = __builtin_nontemporal_load(&target[b + k]);
        unsigned char m = __builtin_nontemporal_load(&mask[b + k]);
        x[k] = m ? p : t;
    }
    float s, c;
    __sincosf(x[5], &s, &c);
    f32x2 v; v.x = x[4] * c; v.y = x[4] * s;
    f32x2 sb; sb.x = x[2] - x[0]; sb.y = x[3] - x[1];
    Vc[i] = v;   // b64 store
    Sc[i] = sb;  // b64 store
}

// Kernel 2b: scalar tail for odd E (edges [base, E)).
__global__ void pbe_edge_tail(const f32x2* __restrict__ attr,
                              const long long* __restrict__ eidx,
                              const f32x2* __restrict__ Vc,
                              float* __restrict__ Ic,
                              int base, int E) {
    int e = base + blockIdx.x * blockDim.x + threadIdx.x;
    if (e >= E) return;
    long long s = eidx[e];
    long long d = eidx[(long long)E + e];
    f32x2 gb = attr[e];
    f32x2 v = Vc[s];
    unsafeAtomicAdd(&Ic[2 * d + 0], gb.x * v.x - gb.y * v.y);
    unsafeAtomicAdd(&Ic[2 * d + 1], gb.x * v.y + gb.y * v.x);
}

// ---------------------------------------------------------------------------
// Kernel 3: residual r = S - V*conj(I); accumulate sum|r|, sum|Re r|, sum|Im r|
// wave32 shuffle reduce -> LDS cross-wave reduce -> f64 global atomics.
// ---------------------------------------------------------------------------
__global__ void pbe_residual(const f32x2* __restrict__ Vc,
                             const f32x2* __restrict__ Sc,
                             const f32x2* __restrict__ Ic,
                             double* __restrict__ acc, int N) {
    int i = blockIdx.x * blockDim.x + threadIdx.x;
    float a0 = 0.f, a1 = 0.f, a2 = 0.f;
    if (i < N) {
        f32x2 v = Vc[i];
        f32x2 sb = Sc[i];
        f32x2 cur = Ic[i];
        // V * conj(I) = (vr*ir + vi*ii) + j(vi*ir - vr*ii)
        float rr  = sb.x - (v.x * cur.x + v.y * cur.y);
        float rim = sb.y - (v.y * cur.x - v.x * cur.y);
        a1 = fabsf(rr);
        a2 = fabsf(rim);
        a0 = sqrtf(rr * rr + rim * rim);
    }
#pragma unroll
    for (int off = 16; off > 0; off >>= 1) {
        a0 += __shfl_down(a0, off);
        a1 += __shfl_down(a1, off);
        a2 += __shfl_down(a2, off);
    }
    __shared__ float s0[THREADS / 32], s1[THREADS / 32], s2[THREADS / 32];
    const int lane = threadIdx.x & 31;
    const int wave = threadIdx.x >> 5;
    if (lane == 0) { s0[wave] = a0; s1[wave] = a1; s2[wave] = a2; }
    __syncthreads();
    if (wave == 0) {
        const int nw = THREADS / 32;  // 8
        a0 = (lane < nw) ? s0[lane] : 0.f;
        a1 = (lane < nw) ? s1[lane] : 0.f;
        a2 = (lane < nw) ? s2[lane] : 0.f;
#pragma unroll
        for (int off = nw / 2; off > 0; off >>= 1) {
            a0 += __shfl_down(a0, off);
            a1 += __shfl_down(a1, off);
            a2 += __shfl_down(a2, off);
        }
        if (lane == 0) {
            unsafeAtomicAdd(&acc[0], (double)a0);  // global_atomic_add_f64
            unsafeAtomicAdd(&acc[1], (double)a1);
            unsafeAtomicAdd(&acc[2], (double)a2);
        }
    }
}

// ---------------------------------------------------------------------------
// Kernel 4: finalize means -> d_out[3] = {loss, real_loss, imag_loss}
// ---------------------------------------------------------------------------
__global__ void pbe_finalize(const double* __restrict__ acc,
                             float* __restrict__ out, int N) {
    int k = threadIdx.x;
    if (k < 3) out[k] = (float)(acc[k] / (double)N);
}

// ---------------------------------------------------------------------------
extern "C" void kernel_launch(void* const* d_in, const int* in_sizes, int n_in,
                              void* d_out, int out_size, void* d_ws, size_t ws_size,
                              hipStream_t stream) {
    const float* pred         = (const float*)d_in[0];
    const float* target       = (const float*)d_in[1];
    const float* edge_attr    = (const float*)d_in[2];
    const long long* eidx     = (const long long*)d_in[3];
    const unsigned char* mask = (const unsigned char*)d_in[4];  // jnp bool_ = 1 byte

    const int N = in_sizes[0] / 6;   // 2,000,000
    const int E = in_sizes[2] / 2;   // 32,000,000

    // Workspace layout: [acc: 3 doubles (+pad to 32B)] [Vc: 2N f32][Sc: 2N][Ic: 2N]
    char* ws = (char*)d_ws;
    double* acc = (double*)ws;
    float* Vc = (float*)(ws + 32);
    float* Sc = Vc + 2 * (size_t)N;
    float* Ic = Sc + 2 * (size_t)N;

    hipMemsetAsync(acc, 0, 32, stream);
    hipMemsetAsync(Ic, 0, (size_t)2 * (size_t)N * sizeof(float), stream);

    pbe_node_prep<<<(N + THREADS - 1) / THREADS, THREADS, 0, stream>>>(
        pred, target, mask, (f32x2*)Vc, (f32x2*)Sc, N);

    const int nPairs = E / 2;
    if (nPairs > 0) {
        pbe_edge_pair<<<(nPairs + THREADS - 1) / THREADS, THREADS, 0, stream>>>(
            (const f32x4*)edge_attr,
            (const i64x2*)eidx,
            (const i64x2*)(eidx + (long long)E),
            (const f32x2*)Vc, Ic, nPairs);
    }
    if (E & 1) {
        pbe_edge_tail<<<1, 1, 0, stream>>>(
            (const f32x2*)edge_attr, eidx, (const f32x2*)Vc, Ic, E - 1, E);
    }

    pbe_residual<<<(N + THREADS - 1) / THREADS, THREADS, 0, stream>>>(
        (const f32x2*)Vc, (const f32x2*)Sc, (const f32x2*)Ic, acc, N);

    pbe_finalize<<<1, 32, 0, stream>>>(acc, (float*)d_out, N);
}